// ColourCatGINConv_41094247088188
// MI455X (gfx1250) — compile-verified
//
#include <hip/hip_runtime.h>

typedef float v2f __attribute__((ext_vector_type(2)));
typedef float v8f __attribute__((ext_vector_type(8)));

#define N_NODES    100000
#define N_EDGES    1600000
#define IN_DIM     128
#define COLOUR_DIM 32
#define EMB_DIM    128
#define BN_EPS     1e-5f

// ---- workspace layout ----
// float region (element offsets from (float*)d_ws)
#define WS_AGG    0              // 100000*128
#define WS_SC     12800000       // 100000
#define WS_DEG    12900000       // 100000
#define WS_CSUM   13000000       // 128
#define WS_CSQ    13000128       // 128
#define WS_U      13000256       // 128
#define WS_V      13000384       // 128
#define WS_SCALE  13000512       // 128
#define WS_SHIFT  13000640       // 128
#define WS_Y      13000768       // 100000*128
// int region starts after float region (element offsets from (int*)base)
#define WS_INT_BASE 25800768     // float elements consumed above
#define WI_CNT    0              // 100001
#define WI_OFF    100016         // 100001 (padded starts for alignment)
#define WI_CUR    200032         // 100001
#define WI_CSR    300048         // 1600000

// Zero CSR counters + BN stat accumulators (runs every call -> deterministic).
__global__ void k_zero(int* __restrict__ cnt, float* __restrict__ stats) {
    int stride = gridDim.x * blockDim.x;
    for (int i = blockIdx.x * blockDim.x + threadIdx.x; i <= N_NODES; i += stride)
        cnt[i] = 0;
    for (int i = blockIdx.x * blockDim.x + threadIdx.x; i < 256; i += stride)
        stats[i] = 0.f;  // csum(128) + csq(128), contiguous
}

// Phase 1: indegree histogram.
__global__ __launch_bounds__(256)
void k_count(const int* __restrict__ ei, int* __restrict__ cnt) {
    int stride = gridDim.x * blockDim.x;
    for (int e = blockIdx.x * blockDim.x + threadIdx.x; e < N_EDGES; e += stride)
        atomicAdd(&cnt[ei[N_EDGES + e]], 1);
}

// Phase 2: exclusive prefix sum over 100000 counts, single 1024-thread block.
// Each thread serially owns CHUNK elements; block-level Hillis-Steele scan.
__global__ __launch_bounds__(1024)
void k_scan(const int* __restrict__ cnt, int* __restrict__ off,
            int* __restrict__ cur) {
    __shared__ int ssum[1024];
    const int CHUNK = (N_NODES + 1023) / 1024;  // 98
    int tid  = threadIdx.x;
    int base = tid * CHUNK;

    int s = 0;
    for (int i = 0; i < CHUNK; ++i) {
        int n = base + i;
        if (n < N_NODES) s += cnt[n];
    }
    ssum[tid] = s;
    __syncthreads();
    for (int d = 1; d < 1024; d <<= 1) {
        int t = 0;
        if (tid >= d) t = ssum[tid - d];
        __syncthreads();
        if (tid >= d) ssum[tid] += t;
        __syncthreads();
    }
    int run = (tid == 0) ? 0 : ssum[tid - 1];  // exclusive offset of this chunk
    for (int i = 0; i < CHUNK; ++i) {
        int n = base + i;
        if (n < N_NODES) {
            off[n] = run;
            cur[n] = run;
            run += cnt[n];
        }
    }
    if (tid == 0) off[N_NODES] = ssum[1023];  // == N_EDGES
}

// Phase 3: bucket src indices by dst.
__global__ __launch_bounds__(256)
void k_fill(const int* __restrict__ ei, int* __restrict__ cur,
            int* __restrict__ csr_src) {
    int stride = gridDim.x * blockDim.x;
    for (int e = blockIdx.x * blockDim.x + threadIdx.x; e < N_EDGES; e += stride) {
        int src = ei[e];
        int dst = ei[N_EDGES + e];
        int pos = atomicAdd(&cur[dst], 1);
        csr_src[pos] = src;
    }
}

// Phase 4: gather-sum. One wave32 per node; lane owns a float4 column slice.
// Zero atomics: agg / s_c / deg each written once by their owning wave.
__global__ __launch_bounds__(256)
void k_gather(const float* __restrict__ x, const float* __restrict__ c,
              const int* __restrict__ off, const int* __restrict__ csr_src,
              float* __restrict__ agg, float* __restrict__ sc,
              float* __restrict__ deg) {
    int lane   = threadIdx.x & 31;
    int wave   = (int)((blockIdx.x * blockDim.x + threadIdx.x) >> 5);
    int nwaves = (int)((gridDim.x * blockDim.x) >> 5);
    for (int n = wave; n < N_NODES; n += nwaves) {
        int beg = off[n], end = off[n + 1];
        float4 acc = {0.f, 0.f, 0.f, 0.f};
        float  scv = 0.f;
        for (int e = beg; e < end; ++e) {
            int src = csr_src[e];
            if (e + 2 < end) {  // global_prefetch_b8 two edges ahead
                int ps = csr_src[e + 2];
                __builtin_prefetch(x + (size_t)ps * IN_DIM + lane * 4, 0, 3);
            }
            float4 vx = ((const float4*)(x + (size_t)src * IN_DIM))[lane];
            acc.x += vx.x; acc.y += vx.y; acc.z += vx.z; acc.w += vx.w;
            scv += c[src];
        }
        ((float4*)(agg + (size_t)n * IN_DIM))[lane] = acc;
        if (lane == 0) {
            sc[n]  = scv;
            deg[n] = (float)(end - beg);
        }
    }
}

// u = colour_W @ W1[128:160,:], v = colour_b @ W1[128:160,:]
__global__ void k_uv(const float* __restrict__ cW, const float* __restrict__ cb,
                     const float* __restrict__ W1, float* __restrict__ u,
                     float* __restrict__ v) {
    int j = threadIdx.x;  // 0..127
    float su = 0.f, sv = 0.f;
    #pragma unroll 8
    for (int t = 0; t < COLOUR_DIM; ++t) {
        float wv = W1[(size_t)(IN_DIM + t) * EMB_DIM + j];
        su = fmaf(cW[t], wv, su);
        sv = fmaf(cb[t], wv, sv);
    }
    u[j] = su;
    v[j] = sv;
}

// GEMM1: y = ((1+eps)*x + agg) @ W1[:128,:] + alpha*u + betan*v + b1,
// fused BN column-statistics. 16x128 tile per block, 8 waves, each wave a
// 16x16 tile via V_WMMA_F32_16X16X4_F32.
__global__ __launch_bounds__(256)
void k_gemm1(const float* __restrict__ x, const float* __restrict__ c,
             const float* __restrict__ eps_p, const float* __restrict__ W1,
             const float* __restrict__ b1, const float* __restrict__ agg,
             const float* __restrict__ sc, const float* __restrict__ deg,
             const float* __restrict__ u, const float* __restrict__ v,
             float* __restrict__ y, float* __restrict__ csum,
             float* __restrict__ csq) {
    __shared__ float zx[16][132];   // pitch 132 breaks 64-bank conflicts
    __shared__ float alpha[16];
    __shared__ float betan[16];

    int tid  = threadIdx.x;
    int row0 = blockIdx.x * 16;
    float se = 1.0f + eps_p[0];

    #pragma unroll
    for (int j = 0; j < 8; ++j) {
        int i = tid + 256 * j;  // 0..2047
        int r = i >> 7, k = i & 127;
        size_t g = (size_t)(row0 + r) * IN_DIM + k;
        zx[r][k] = fmaf(se, x[g], agg[g]);
    }
    if (tid < 16) {
        int n = row0 + tid;
        alpha[tid] = fmaf(se, c[n], sc[n]);  // (1+eps)*c + sum_c
        betan[tid] = se + deg[n];            // (1+eps) + indegree
    }
    __syncthreads();

    int wv = tid >> 5, lane = tid & 31;
    int col0  = wv * 16;
    int nn    = lane & 15;          // A row / B col for this lane
    int khalf = (lane >> 4) * 2;    // K sub-pair selector

    v8f acc = {};
    #pragma unroll 4
    for (int k0 = 0; k0 < IN_DIM; k0 += 4) {
        v2f a, b;
        a.x = zx[nn][k0 + khalf];
        a.y = zx[nn][k0 + khalf + 1];
        const float* Wp = W1 + (size_t)(k0 + khalf) * EMB_DIM + col0 + nn;
        b.x = Wp[0];
        b.y = Wp[EMB_DIM];
        acc = __builtin_amdgcn_wmma_f32_16x16x4_f32(
            false, a, false, b, (short)0, acc, false, false);
    }

    float ucol = u[col0 + nn], vcol = v[col0 + nn], bcol = b1[col0 + nn];
    int mbase = (lane >> 4) * 8;
    float s1 = 0.f, s2 = 0.f;
    #pragma unroll
    for (int r = 0; r < 8; ++r) {
        int m = mbase + r;
        float val = acc[r] + alpha[m] * ucol + betan[m] * vcol + bcol;
        y[(size_t)(row0 + m) * EMB_DIM + col0 + nn] = val;
        s1 += val;
        s2 = fmaf(val, val, s2);
    }
    atomicAdd(&csum[col0 + nn], s1);
    atomicAdd(&csq[col0 + nn], s2);
}

// Fold BN statistics into per-column scale/shift.
__global__ void k_bn(const float* __restrict__ csum, const float* __restrict__ csq,
                     const float* __restrict__ gamma, const float* __restrict__ beta,
                     float* __restrict__ scale, float* __restrict__ shift) {
    int j = threadIdx.x;  // 0..127
    float inv = 1.0f / (float)N_NODES;
    float mu  = csum[j] * inv;
    float var = csq[j] * inv - mu * mu;
    float rs  = rsqrtf(var + BN_EPS);
    float s   = gamma[j] * rs;
    scale[j] = s;
    shift[j] = fmaf(-mu, s, beta[j]);
}

// GEMM2: out = relu(y*scale + shift) @ W2 + b2 (BN+ReLU fused into A staging).
__global__ __launch_bounds__(256)
void k_gemm2(const float* __restrict__ y, const float* __restrict__ scale,
             const float* __restrict__ shift, const float* __restrict__ W2,
             const float* __restrict__ b2, float* __restrict__ out) {
    __shared__ float at[16][132];
    int tid  = threadIdx.x;
    int row0 = blockIdx.x * 16;

    #pragma unroll
    for (int j = 0; j < 8; ++j) {
        int i = tid + 256 * j;
        int r = i >> 7, k = i & 127;
        float val = fmaf(y[(size_t)(row0 + r) * EMB_DIM + k], scale[k], shift[k]);
        at[r][k] = fmaxf(val, 0.0f);
    }
    __syncthreads();

    int wv = tid >> 5, lane = tid & 31;
    int col0  = wv * 16;
    int nn    = lane & 15;
    int khalf = (lane >> 4) * 2;

    v8f acc = {};
    #pragma unroll 4
    for (int k0 = 0; k0 < EMB_DIM; k0 += 4) {
        v2f a, b;
        a.x = at[nn][k0 + khalf];
        a.y = at[nn][k0 + khalf + 1];
        const float* Wp = W2 + (size_t)(k0 + khalf) * EMB_DIM + col0 + nn;
        b.x = Wp[0];
        b.y = Wp[EMB_DIM];
        acc = __builtin_amdgcn_wmma_f32_16x16x4_f32(
            false, a, false, b, (short)0, acc, false, false);
    }

    float bc = b2[col0 + nn];
    int mbase = (lane >> 4) * 8;
    #pragma unroll
    for (int r = 0; r < 8; ++r)
        out[(size_t)(row0 + mbase + r) * EMB_DIM + col0 + nn] = acc[r] + bc;
}

extern "C" void kernel_launch(void* const* d_in, const int* in_sizes, int n_in,
                              void* d_out, int out_size, void* d_ws, size_t ws_size,
                              hipStream_t stream) {
    const float* x     = (const float*)d_in[0];
    const float* c     = (const float*)d_in[1];
    const int*   ei    = (const int*)d_in[2];
    const float* cW    = (const float*)d_in[3];
    const float* cb    = (const float*)d_in[4];
    const float* eps   = (const float*)d_in[5];
    const float* W1    = (const float*)d_in[6];
    const float* b1    = (const float*)d_in[7];
    const float* gamma = (const float*)d_in[8];
    const float* beta  = (const float*)d_in[9];
    const float* W2    = (const float*)d_in[10];
    const float* b2    = (const float*)d_in[11];

    float* w    = (float*)d_ws;
    int*   wi   = (int*)(w + WS_INT_BASE);
    float* out  = (float*)d_out;

    int* cnt = wi + WI_CNT;
    int* off = wi + WI_OFF;
    int* cur = wi + WI_CUR;
    int* csr = wi + WI_CSR;

    k_zero<<<256, 256, 0, stream>>>(cnt, w + WS_CSUM);
    k_uv<<<1, 128, 0, stream>>>(cW, cb, W1, w + WS_U, w + WS_V);
    k_count<<<2048, 256, 0, stream>>>(ei, cnt);
    k_scan<<<1, 1024, 0, stream>>>(cnt, off, cur);
    k_fill<<<2048, 256, 0, stream>>>(ei, cur, csr);
    k_gather<<<4096, 256, 0, stream>>>(x, c, off, csr, w + WS_AGG, w + WS_SC,
                                       w + WS_DEG);
    k_gemm1<<<N_NODES / 16, 256, 0, stream>>>(x, c, eps, W1, b1, w + WS_AGG,
                                              w + WS_SC, w + WS_DEG, w + WS_U,
                                              w + WS_V, w + WS_Y, w + WS_CSUM,
                                              w + WS_CSQ);
    k_bn<<<1, 128, 0, stream>>>(w + WS_CSUM, w + WS_CSQ, gamma, beta,
                                w + WS_SCALE, w + WS_SHIFT);
    k_gemm2<<<N_NODES / 16, 256, 0, stream>>>(w + WS_Y, w + WS_SCALE, w + WS_SHIFT,
                                              W2, b2, out);
}